// RADS_79912161509980
// MI455X (gfx1250) — compile-verified
//
#include <hip/hip_runtime.h>
#include <hip/hip_bf16.h>
#include <math.h>

typedef __attribute__((ext_vector_type(2))) float v2f;
typedef __attribute__((ext_vector_type(8))) float v8f;

#define NB 256          // batch
#define NN 200000       // library size
#define DD 64           // feature dim
#define HH 64           // hidden
#define ACTD 6
#define SPAN 2048       // N columns per block in WMMA passes
#define NTILE 16        // 128-col tiles per block (16*128 = 2048)

static __device__ __forceinline__ v8f wmma4(v2f a, v2f b, v8f c) {
    // D = A(16x4 f32) * B(4x16 f32) + C(16x16 f32)
    return __builtin_amdgcn_wmma_f32_16x16x4_f32(false, a, false, b, (short)0, c, false, false);
}

static __device__ __forceinline__ void atomicMaxF(float* addr, float val) {
    int* ia = (int*)addr;
    int old = __float_as_int(*addr);
    while (val > __int_as_float(old)) {
        int assumed = old;
        old = atomicCAS(ia, assumed, __float_as_int(val));
        if (old == assumed) break;
    }
}

// ---------------- K0: init workspace ----------------
__global__ __launch_bounds__(256) void k_init(float* rowmax, float* rowsum, float* ctx) {
    int i = blockIdx.x * 256 + threadIdx.x;
    if (i < NB) { rowmax[i] = -INFINITY; rowsum[i] = 0.0f; }
    if (i < NB * DD) ctx[i] = 0.0f;
}

// ---------------- K1: same-waypoint valid count per row ----------------
__global__ __launch_bounds__(256) void k_count(const unsigned char* __restrict__ mask,
                                               const int* __restrict__ lib_wp,
                                               const int* __restrict__ wp_id,
                                               int* __restrict__ cnt) {
    __shared__ int red[256];
    const int b = blockIdx.x;
    const int wpb = wp_id[b];
    const unsigned char* mrow = mask + (size_t)b * NN;
    int c = 0;
    for (int i0 = threadIdx.x * 16; i0 < NN; i0 += 256 * 16) {
        uint4 mv = *(const uint4*)(mrow + i0);
        unsigned mm[4] = { mv.x, mv.y, mv.z, mv.w };
#pragma unroll
        for (int qd = 0; qd < 4; qd++) {
            int4 w = *(const int4*)(lib_wp + i0 + qd * 4);
            unsigned mb = mm[qd];
            c += (((mb      ) & 0xFF) == 0) & (w.x == wpb);
            c += (((mb >>  8) & 0xFF) == 0) & (w.y == wpb);
            c += (((mb >> 16) & 0xFF) == 0) & (w.z == wpb);
            c += (((mb >> 24) & 0xFF) == 0) & (w.w == wpb);
        }
    }
    red[threadIdx.x] = c;
    __syncthreads();
    for (int s = 128; s > 0; s >>= 1) {
        if (threadIdx.x < s) red[threadIdx.x] += red[threadIdx.x + s];
        __syncthreads();
    }
    if (threadIdx.x == 0) cnt[b] = red[0];
}

// ---------------- K2: WMMA score pass -> per-row global max ----------------
__global__ __launch_bounds__(256) void k_scoremax(const float* __restrict__ q,
                                                  const float* __restrict__ lib,
                                                  const unsigned char* __restrict__ mask,
                                                  const int* __restrict__ lib_wp,
                                                  const int* __restrict__ wp_id,
                                                  const int* __restrict__ cnt,
                                                  const float* __restrict__ log_tau_p,
                                                  float* __restrict__ rowmax) {
    __shared__ float q_lds[16][66];              // padded: stride 66 kills bank conflicts
    __shared__ unsigned long long mask_lds[16][16]; // 16 rows x 128 bytes
    __shared__ int wp_lds[16];
    __shared__ int suff_lds[16];
    __shared__ float blockmax[16];

    const int tid  = threadIdx.x;
    const int wave = tid >> 5;
    const int lane = tid & 31;
    const int half = lane >> 4;
    const int m    = lane & 15;
    const int r0   = blockIdx.y * 16;

    for (int i = tid; i < 16 * 64; i += 256)
        q_lds[i >> 6][i & 63] = q[(size_t)(r0 + (i >> 6)) * DD + (i & 63)];
    if (tid < 16) {
        wp_lds[tid]   = wp_id[r0 + tid];
        suff_lds[tid] = (cnt[r0 + tid] >= 8);
        blockmax[tid] = -INFINITY;
    }
    __syncthreads();

    const float inv_tau = 1.0f / fminf(fmaxf(__expf(log_tau_p[0]), 1e-3f), 10.0f);

    float mx[8];
#pragma unroll
    for (int j = 0; j < 8; j++) mx[j] = -INFINITY;

    const int span0 = blockIdx.x * SPAN;
    for (int t = 0; t < NTILE; ++t) {
        const int nt = span0 + t * 128;
        if (nt >= NN) break;                       // uniform across block
        {   // stage 16x128 mask bytes
            int row = tid >> 4, seg = tid & 15;
            int c0 = nt + seg * 8;
            unsigned long long v = ~0ULL;
            if (c0 < NN) v = *(const unsigned long long*)(mask + (size_t)(r0 + row) * NN + c0);
            mask_lds[row][seg] = v;
        }
        __syncthreads();

        const int nb = nt + wave * 16;
        if (nb < NN) {
            v8f acc = {0.f,0.f,0.f,0.f,0.f,0.f,0.f,0.f};
            const float* lrow = lib + (size_t)(nb + m) * DD;
            const float* qrow = &q_lds[m][0];
#pragma unroll
            for (int k = 0; k < 16; k++) {
                const int kk = 4 * k + 2 * half;
                v2f a, b;
                a[0] = qrow[kk]; a[1] = qrow[kk + 1];
                b[0] = lrow[kk]; b[1] = lrow[kk + 1];
                acc = wmma4(a, b, acc);
            }
            const int wpcol = lib_wp[nb + m];
#pragma unroll
            for (int j = 0; j < 8; j++) {
                const int M = j + 8 * half;
                float s = acc[j] * inv_tau;
                unsigned char mb = ((const unsigned char*)mask_lds[M])[wave * 16 + m];
                bool kill = mb || (suff_lds[M] && (wpcol != wp_lds[M]));
                s = kill ? -INFINITY : s;
                mx[j] = fmaxf(mx[j], s);
            }
        }
        __syncthreads();
    }
#pragma unroll
    for (int j = 0; j < 8; j++) {
        float v = mx[j];
        for (int o = 8; o > 0; o >>= 1) v = fmaxf(v, __shfl_xor(v, o, 32));
        if (m == 0) atomicMaxF(&blockmax[j + 8 * half], v);
    }
    __syncthreads();
    if (tid < 16) atomicMaxF(&rowmax[r0 + tid], blockmax[tid]);
}

// ---------------- K3: p=exp(s-max), write attn, rowsum, WMMA context ----------------
__global__ __launch_bounds__(256) void k_softmax_ctx(const float* __restrict__ q,
                                                     const float* __restrict__ lib,
                                                     const unsigned char* __restrict__ mask,
                                                     const int* __restrict__ lib_wp,
                                                     const int* __restrict__ wp_id,
                                                     const int* __restrict__ cnt,
                                                     const float* __restrict__ log_tau_p,
                                                     const float* __restrict__ rowmax,
                                                     float* __restrict__ rowsum,
                                                     float* __restrict__ ctx,
                                                     float* __restrict__ attn) {
    __shared__ float q_lds[16][66];
    __shared__ unsigned long long mask_lds[16][16];
    __shared__ int wp_lds[16];
    __shared__ int suff_lds[16];
    __shared__ float rmax_lds[16];
    __shared__ float blocksum[16];
    __shared__ float p_lds[8][16][17];           // per-wave P tile, padded
    __shared__ float ctx_block[16][64];

    const int tid  = threadIdx.x;
    const int wave = tid >> 5;
    const int lane = tid & 31;
    const int half = lane >> 4;
    const int m    = lane & 15;
    const int r0   = blockIdx.y * 16;

    for (int i = tid; i < 16 * 64; i += 256) {
        q_lds[i >> 6][i & 63] = q[(size_t)(r0 + (i >> 6)) * DD + (i & 63)];
        ctx_block[i >> 6][i & 63] = 0.0f;
    }
    if (tid < 16) {
        wp_lds[tid]   = wp_id[r0 + tid];
        suff_lds[tid] = (cnt[r0 + tid] >= 8);
        rmax_lds[tid] = rowmax[r0 + tid];
        blocksum[tid] = 0.0f;
    }
    __syncthreads();

    const float inv_tau = 1.0f / fminf(fmaxf(__expf(log_tau_p[0]), 1e-3f), 10.0f);

    float psum[8];
    v8f cacc[4];
#pragma unroll
    for (int j = 0; j < 8; j++) psum[j] = 0.0f;
#pragma unroll
    for (int d = 0; d < 4; d++) cacc[d] = (v8f){0.f,0.f,0.f,0.f,0.f,0.f,0.f,0.f};

    const int span0 = blockIdx.x * SPAN;
    for (int t = 0; t < NTILE; ++t) {
        const int nt = span0 + t * 128;
        if (nt >= NN) break;
        {
            int row = tid >> 4, seg = tid & 15;
            int c0 = nt + seg * 8;
            unsigned long long v = ~0ULL;
            if (c0 < NN) v = *(const unsigned long long*)(mask + (size_t)(r0 + row) * NN + c0);
            mask_lds[row][seg] = v;
        }
        __syncthreads();

        const int nb = nt + wave * 16;
        if (nb < NN) {
            v8f acc = {0.f,0.f,0.f,0.f,0.f,0.f,0.f,0.f};
            const float* lrow = lib + (size_t)(nb + m) * DD;
            const float* qrow = &q_lds[m][0];
#pragma unroll
            for (int k = 0; k < 16; k++) {
                const int kk = 4 * k + 2 * half;
                v2f a, b;
                a[0] = qrow[kk]; a[1] = qrow[kk + 1];
                b[0] = lrow[kk]; b[1] = lrow[kk + 1];
                acc = wmma4(a, b, acc);
            }
            const int wpcol = lib_wp[nb + m];
#pragma unroll
            for (int j = 0; j < 8; j++) {
                const int M = j + 8 * half;
                float s = acc[j] * inv_tau;
                unsigned char mb = ((const unsigned char*)mask_lds[M])[wave * 16 + m];
                bool kill = mb || (suff_lds[M] && (wpcol != wp_lds[M]));
                float p = kill ? 0.0f : expf(s - rmax_lds[M]);
                psum[j] += p;
                attn[(size_t)(r0 + M) * NN + nb + m] = p;   // unnormalized, fixed in K5
                p_lds[wave][M][m] = p;
            }
            // context: ctx[M][d] += P(16xK=16) * Lib(K=16 x 16), via 4 k-steps x 4 d-tiles
#pragma unroll
            for (int d0 = 0; d0 < 4; d0++) {
#pragma unroll
                for (int koff = 0; koff < 16; koff += 4) {
                    const int kk = koff + 2 * half;
                    v2f a, b;
                    a[0] = p_lds[wave][m][kk];
                    a[1] = p_lds[wave][m][kk + 1];
                    b[0] = lib[(size_t)(nb + kk)     * DD + d0 * 16 + m];
                    b[1] = lib[(size_t)(nb + kk + 1) * DD + d0 * 16 + m];
                    cacc[d0] = wmma4(a, b, cacc[d0]);
                }
            }
        }
        __syncthreads();
    }

#pragma unroll
    for (int j = 0; j < 8; j++) {
        float v = psum[j];
        for (int o = 8; o > 0; o >>= 1) v += __shfl_xor(v, o, 32);
        if (m == 0) atomicAdd(&blocksum[j + 8 * half], v);
    }
#pragma unroll
    for (int d0 = 0; d0 < 4; d0++)
#pragma unroll
        for (int j = 0; j < 8; j++)
            atomicAdd(&ctx_block[j + 8 * half][d0 * 16 + m], cacc[d0][j]);
    __syncthreads();
    if (tid < 16) atomicAdd(&rowsum[r0 + tid], blocksum[tid]);
    for (int i = tid; i < 16 * 64; i += 256)
        atomicAdd(&ctx[(size_t)(r0 + (i >> 6)) * DD + (i & 63)], ctx_block[i >> 6][i & 63]);
}

// ---------------- K4: normalize context + GELU MLP + prior logit ----------------
__global__ __launch_bounds__(256) void k_mlp(const float* __restrict__ q,
                                             const float* __restrict__ W1,
                                             const float* __restrict__ b1,
                                             const float* __restrict__ W2,
                                             const float* __restrict__ b2,
                                             const float* __restrict__ action_emb,
                                             const int* __restrict__ action_id,
                                             const float* __restrict__ prior,
                                             const float* __restrict__ plw_p,
                                             const float* __restrict__ rowsum,
                                             const float* __restrict__ ctx,
                                             float* __restrict__ out_logits) {
    __shared__ float w1s[(2 * DD + ACTD) * HH];   // 134x64 = 34.3KB
    const int tid = threadIdx.x;
    for (int i = tid; i < (2 * DD + ACTD) * HH; i += 256) w1s[i] = W1[i];
    __syncthreads();

    const int b = tid;
    float h[2 * DD + ACTD];
    const float inv = 1.0f / rowsum[b];
#pragma unroll
    for (int i = 0; i < DD; i++) h[i] = q[(size_t)b * DD + i];
#pragma unroll
    for (int i = 0; i < DD; i++) h[DD + i] = ctx[(size_t)b * DD + i] * inv;
    const int a = action_id[b];
#pragma unroll
    for (int i = 0; i < ACTD; i++) h[2 * DD + i] = action_emb[a * ACTD + i];

    float acc2 = b2[0];
    for (int j = 0; j < HH; j++) {
        float s = b1[j];
        for (int i = 0; i < 2 * DD + ACTD; i++) s += h[i] * w1s[i * HH + j];
        float g = 0.5f * s * (1.0f + erff(s * 0.70710678118654752f));
        acc2 += g * W2[j];
    }
    float p = fminf(fmaxf(prior[b], 1e-3f), 1.0f - 1e-3f);
    out_logits[b] = acc2 + plw_p[0] * (logf(p) - log1pf(-p));
}

// ---------------- K5: normalize attn in place ----------------
__global__ __launch_bounds__(256) void k_norm(const float* __restrict__ rowsum,
                                              float* __restrict__ attn) {
    const int b = blockIdx.y;
    const int i = blockIdx.x * 256 + threadIdx.x;
    if (i < NN / 4) {
        const float inv = 1.0f / rowsum[b];
        float4* p = (float4*)(attn + (size_t)b * NN) + i;
        float4 v = *p;
        v.x *= inv; v.y *= inv; v.z *= inv; v.w *= inv;
        *p = v;
    }
}

extern "C" void kernel_launch(void* const* d_in, const int* in_sizes, int n_in,
                              void* d_out, int out_size, void* d_ws, size_t ws_size,
                              hipStream_t stream) {
    const float*         q        = (const float*)d_in[0];
    const float*         lib      = (const float*)d_in[1];
    const unsigned char* mask     = (const unsigned char*)d_in[2];
    const int*           lib_wp   = (const int*)d_in[3];
    const int*           wp_id    = (const int*)d_in[4];
    const int*           act_id   = (const int*)d_in[5];
    const float*         prior    = (const float*)d_in[6];
    const float*         act_emb  = (const float*)d_in[7];
    const float*         W1       = (const float*)d_in[8];
    const float*         b1      = (const float*)d_in[9];
    const float*         W2       = (const float*)d_in[10];
    const float*         b2      = (const float*)d_in[11];
    const float*         log_tau  = (const float*)d_in[12];
    const float*         plw      = (const float*)d_in[13];

    float* out    = (float*)d_out;
    float* logits = out;            // [256]
    float* attn   = out + NB;       // [256 x 200000]

    float* wsf    = (float*)d_ws;
    float* rowmax = wsf;            // 256
    float* rowsum = wsf + 256;      // 256
    int*   cnt    = (int*)(wsf + 512); // 256
    float* ctx    = wsf + 768;      // 256*64

    const int nxblocks = (NN + SPAN - 1) / SPAN; // 98

    hipLaunchKernelGGL(k_init, dim3((NB * DD + 255) / 256), dim3(256), 0, stream,
                       rowmax, rowsum, ctx);
    hipLaunchKernelGGL(k_count, dim3(NB), dim3(256), 0, stream,
                       mask, lib_wp, wp_id, cnt);
    hipLaunchKernelGGL(k_scoremax, dim3(nxblocks, NB / 16), dim3(256), 0, stream,
                       q, lib, mask, lib_wp, wp_id, cnt, log_tau, rowmax);
    hipLaunchKernelGGL(k_softmax_ctx, dim3(nxblocks, NB / 16), dim3(256), 0, stream,
                       q, lib, mask, lib_wp, wp_id, cnt, log_tau, rowmax, rowsum, ctx, attn);
    hipLaunchKernelGGL(k_mlp, dim3(1), dim3(256), 0, stream,
                       q, W1, b1, W2, b2, act_emb, act_id, prior, plw, rowsum, ctx, logits);
    hipLaunchKernelGGL(k_norm, dim3((NN / 4 + 255) / 256, NB), dim3(256), 0, stream,
                       rowsum, attn);
}